// MyLSTM_57690000720364
// MI455X (gfx1250) — compile-verified
//
#include <hip/hip_runtime.h>

typedef float v2f __attribute__((ext_vector_type(2)));
typedef float v8f __attribute__((ext_vector_type(8)));

#define BATCH   128
#define HID     512
#define TSEQ    512
#define NWG     16          // workgroups; each owns HB columns of H
#define HB      (HID/NWG)   // 32
#define THREADS 256         // 8 wave32 per WG

__device__ __forceinline__ float fsig(float x)  { return 1.0f / (1.0f + __expf(-x)); }
__device__ __forceinline__ float ftanhf(float x){ return 2.0f / (1.0f + __expf(-2.0f*x)) - 1.0f; }

// ---- monotonic-counter grid barrier (counter reset by init kernel each launch) ----
__device__ __forceinline__ void grid_barrier(unsigned* cnt, unsigned* epoch) {
    __syncthreads();
    if (threadIdx.x == 0) {
        __threadfence();
        __hip_atomic_fetch_add(cnt, 1u, __ATOMIC_RELEASE, __HIP_MEMORY_SCOPE_AGENT);
        const unsigned target = (*epoch) * gridDim.x;
        while (__hip_atomic_load(cnt, __ATOMIC_ACQUIRE, __HIP_MEMORY_SCOPE_AGENT) < target) {
            __builtin_amdgcn_s_sleep(2);
        }
        __threadfence();
    }
    __syncthreads();
    ++(*epoch);
}

#define WMMA_F32(a, b, c) __builtin_amdgcn_wmma_f32_16x16x4_f32( \
        false, (a), false, (b), (short)0, (c), false, false)

// acc[nt] (+)= A_tile x W_tile over K=HID using fp32 WMMA 16x16x4.
// Manually software-pipelined (2 stages), K-loop kept rolled (#pragma unroll 1)
// so the compiler cannot flatten it and spill the B fragments to scratch.
// INIT=true: first k-group writes acc with an inline-0 C operand (no mov-zero block).
template <bool INIT>
__device__ __forceinline__ void gemm_acc(const float* __restrict__ hsrc, int arow,
                                         const float* __restrict__ W,
                                         const int* __restrict__ offs,
                                         v8f acc[8]) {
    // prologue: k-groups 0 and 4
    v2f a0 = *(const v2f*)(hsrc + arow);
    v2f b0[8];
#pragma unroll
    for (int nt = 0; nt < 8; ++nt) b0[nt] = *(const v2f*)(W + offs[nt]);
    v2f a1 = *(const v2f*)(hsrc + arow + 4);
    v2f b1[8];
#pragma unroll
    for (int nt = 0; nt < 8; ++nt) b1[nt] = *(const v2f*)(W + offs[nt] + 4);

    // peeled k=0 group: constant-zero C folds to the inline-0 WMMA operand
    if (INIT) {
        const v8f zero = {0.f, 0.f, 0.f, 0.f, 0.f, 0.f, 0.f, 0.f};
#pragma unroll
        for (int nt = 0; nt < 8; ++nt) acc[nt] = WMMA_F32(a0, b0[nt], zero);
    } else {
#pragma unroll
        for (int nt = 0; nt < 8; ++nt) acc[nt] = WMMA_F32(a0, b0[nt], acc[nt]);
    }
    // preload k=8, then consume k=4
    a0 = *(const v2f*)(hsrc + arow + 8);
#pragma unroll
    for (int nt = 0; nt < 8; ++nt) b0[nt] = *(const v2f*)(W + offs[nt] + 8);
#pragma unroll
    for (int nt = 0; nt < 8; ++nt) acc[nt] = WMMA_F32(a1, b1[nt], acc[nt]);

#pragma unroll 1
    for (int k = 8; k < HID - 8; k += 8) {
        // stage 1 loads (k+4) in flight over stage 0 WMMAs (k)
        a1 = *(const v2f*)(hsrc + arow + k + 4);
#pragma unroll
        for (int nt = 0; nt < 8; ++nt) b1[nt] = *(const v2f*)(W + offs[nt] + k + 4);
#pragma unroll
        for (int nt = 0; nt < 8; ++nt) acc[nt] = WMMA_F32(a0, b0[nt], acc[nt]);

        // stage 0 loads (k+8) in flight over stage 1 WMMAs (k+4)
        a0 = *(const v2f*)(hsrc + arow + k + 8);
#pragma unroll
        for (int nt = 0; nt < 8; ++nt) b0[nt] = *(const v2f*)(W + offs[nt] + k + 8);
#pragma unroll
        for (int nt = 0; nt < 8; ++nt) acc[nt] = WMMA_F32(a1, b1[nt], acc[nt]);
    }
    // epilogue: k-groups HID-8 (in a0/b0) and HID-4
    {
        const int k = HID - 8;
        a1 = *(const v2f*)(hsrc + arow + k + 4);
#pragma unroll
        for (int nt = 0; nt < 8; ++nt) b1[nt] = *(const v2f*)(W + offs[nt] + k + 4);
#pragma unroll
        for (int nt = 0; nt < 8; ++nt) acc[nt] = WMMA_F32(a0, b0[nt], acc[nt]);
#pragma unroll
        for (int nt = 0; nt < 8; ++nt) acc[nt] = WMMA_F32(a1, b1[nt], acc[nt]);
    }
}

__global__ void lstm_init_ws(unsigned* cnt) {
    if (threadIdx.x == 0) *cnt = 0u;
}

__global__ __launch_bounds__(THREADS, 1)
void lstm_persistent(const float* __restrict__ x,
                     const float* __restrict__ Wih1, const float* __restrict__ Whh1,
                     const float* __restrict__ bih1, const float* __restrict__ bhh1,
                     const float* __restrict__ Wih2, const float* __restrict__ Whh2,
                     const float* __restrict__ bih2, const float* __restrict__ bhh2,
                     const float* __restrict__ Wout, const float* __restrict__ bout,
                     const int*   __restrict__ futp,
                     float* __restrict__ out,
                     float* __restrict__ h1buf,   // 2 * B * H (ping-pong)
                     float* __restrict__ h2buf,   // 2 * B * H (ping-pong)
                     float* __restrict__ partials,// NWG * B
                     unsigned* cnt)
{
    __shared__ float c1s[BATCH * HB];
    __shared__ float c2s[BATCH * HB];
    __shared__ float h2s[BATCH * HB];
    __shared__ float x_lds[BATCH];
    __shared__ float out_lds[BATCH];

    const int wg      = blockIdx.x;
    const int tid     = threadIdx.x;
    const int lane    = tid & 31;
    const int wave    = tid >> 5;
    const int lanelow = lane & 15;
    const int hi      = lane >> 4;
    const int m0      = wave * 16;      // B-row tile base for this wave
    const int hb0     = wg * HB;        // H-column base for this WG
    const int khi     = 2 * hi;         // K sub-offset for fp32 WMMA layout
    const int future  = *futp;
    const int TT      = TSEQ + future;

    // ---- zero local cell states and our slices of the global h ping-pong buffers ----
    for (int i = tid; i < BATCH * HB; i += THREADS) { c1s[i] = 0.f; c2s[i] = 0.f; }
    for (int i = tid; i < BATCH * HB; i += THREADS) {
        const int b  = i >> 5;
        const int lh = i & 31;
        h1buf[b * HID + hb0 + lh]               = 0.f;
        h1buf[BATCH * HID + b * HID + hb0 + lh] = 0.f;
        h2buf[b * HID + hb0 + lh]               = 0.f;
        h2buf[BATCH * HID + b * HID + hb0 + lh] = 0.f;
    }
    unsigned epoch = 1;
    grid_barrier(cnt, &epoch);

    // ---- per-lane loop-invariant gate metadata ----
    // N-tile nt covers gate (nt>>1), columns hb0 + (nt&1)*16 + lanelow.
    // Weight row offset n*HID + khi is shared by Whh1 / Wih2 / Whh2 (all [4H, 512]).
    int offs[8];
    float bias1[8], bias2[8], wih[8];
#pragma unroll
    for (int nt = 0; nt < 8; ++nt) {
        const int n = (nt >> 1) * HID + hb0 + (nt & 1) * 16 + lanelow; // gate row in [0,4H)
        offs[nt]  = n * HID + khi;
        bias1[nt] = bih1[n] + bhh1[n];
        bias2[nt] = bih2[n] + bhh2[n];
        wih[nt]   = Wih1[n];            // F==1: x-path of cell 1 is rank-1
    }
    const int   arow = (m0 + lanelow) * HID + khi;   // A-fragment row offset (invariant)
    const float bo   = bout[0];

    int p1 = 0, p2 = 0;
#pragma unroll 1
    for (int t = 0; t < TT; ++t) {
        // stage input (ground truth for t<T, fed-back head output afterwards)
        if (tid < BATCH) x_lds[tid] = (t < TSEQ) ? x[tid * TSEQ + t] : out_lds[tid];
        __syncthreads();

        // =============== cell 1: gates = x (x) Wih1 + h1 @ Whh1^T + b ===============
        v8f acc[8];
        const float* h1cur = h1buf + (size_t)p1 * BATCH * HID;
        gemm_acc<true>(h1cur, arow, Whh1, offs, acc);

        float* h1nxt = h1buf + (size_t)(p1 ^ 1) * BATCH * HID;
#pragma unroll
        for (int r = 0; r < 8; ++r) {
            const int brow = m0 + r + 8 * hi;
            const float xb = x_lds[brow];
#pragma unroll
            for (int p = 0; p < 2; ++p) {
                const int lh = p * 16 + lanelow;
                const float ig = fsig  (acc[0 + p][r] + bias1[0 + p] + xb * wih[0 + p]);
                const float fg = fsig  (acc[2 + p][r] + bias1[2 + p] + xb * wih[2 + p]);
                const float gg = ftanhf(acc[4 + p][r] + bias1[4 + p] + xb * wih[4 + p]);
                const float og = fsig  (acc[6 + p][r] + bias1[6 + p] + xb * wih[6 + p]);
                const float c  = fg * c1s[brow * HB + lh] + ig * gg;
                c1s[brow * HB + lh] = c;
                h1nxt[(size_t)brow * HID + hb0 + lh] = og * ftanhf(c);
            }
        }
        grid_barrier(cnt, &epoch);   // h1nxt fully published

        // =============== cell 2: gates = h1 @ Wih2^T + h2 @ Whh2^T + b ===============
        const float* h2cur = h2buf + (size_t)p2 * BATCH * HID;
        gemm_acc<true >(h1nxt, arow, Wih2, offs, acc);
        gemm_acc<false>(h2cur, arow, Whh2, offs, acc);

        float* h2nxt = h2buf + (size_t)(p2 ^ 1) * BATCH * HID;
#pragma unroll
        for (int r = 0; r < 8; ++r) {
            const int brow = m0 + r + 8 * hi;
#pragma unroll
            for (int p = 0; p < 2; ++p) {
                const int lh = p * 16 + lanelow;
                const float ig = fsig  (acc[0 + p][r] + bias2[0 + p]);
                const float fg = fsig  (acc[2 + p][r] + bias2[2 + p]);
                const float gg = ftanhf(acc[4 + p][r] + bias2[4 + p]);
                const float og = fsig  (acc[6 + p][r] + bias2[6 + p]);
                const float c  = fg * c2s[brow * HB + lh] + ig * gg;
                c2s[brow * HB + lh] = c;
                const float h  = og * ftanhf(c);
                h2nxt[(size_t)brow * HID + hb0 + lh] = h;
                h2s[brow * HB + lh] = h;
            }
        }
        __syncthreads();

        // per-WG partial of the 1-wide head: partial[b] = sum_lh h2[b, hb0+lh] * Wout[hb0+lh]
        if (tid < BATCH) {
            float s = 0.f;
#pragma unroll 8
            for (int lh = 0; lh < HB; ++lh) s += h2s[tid * HB + lh] * Wout[hb0 + lh];
            partials[wg * BATCH + tid] = s;
        }
        grid_barrier(cnt, &epoch);   // all partials + h2nxt published

        // every WG redundantly reduces (deterministic); WG0 also writes the output
        if (tid < BATCH) {
            float s = bo;
#pragma unroll
            for (int j = 0; j < NWG; ++j) s += partials[j * BATCH + tid];
            out_lds[tid] = s;
            if (wg == 0) out[(size_t)tid * TT + t] = s;
        }

        p1 ^= 1; p2 ^= 1;
    }
}

extern "C" void kernel_launch(void* const* d_in, const int* in_sizes, int n_in,
                              void* d_out, int out_size, void* d_ws, size_t ws_size,
                              hipStream_t stream) {
    const float* x    = (const float*)d_in[0];
    const float* Wih1 = (const float*)d_in[1];
    const float* Whh1 = (const float*)d_in[2];
    const float* bih1 = (const float*)d_in[3];
    const float* bhh1 = (const float*)d_in[4];
    const float* Wih2 = (const float*)d_in[5];
    const float* Whh2 = (const float*)d_in[6];
    const float* bih2 = (const float*)d_in[7];
    const float* bhh2 = (const float*)d_in[8];
    const float* Wout = (const float*)d_in[9];
    const float* bout = (const float*)d_in[10];
    const int*   fut  = (const int*)  d_in[11];

    float* out = (float*)d_out;

    // workspace layout: [counter pad 256B][h1 ping-pong][h2 ping-pong][partials]
    char*     ws       = (char*)d_ws;
    unsigned* cnt      = (unsigned*)ws;
    float*    h1buf    = (float*)(ws + 256);
    float*    h2buf    = h1buf + 2 * BATCH * HID;
    float*    partials = h2buf + 2 * BATCH * HID;

    lstm_init_ws<<<1, 64, 0, stream>>>(cnt);
    lstm_persistent<<<NWG, THREADS, 0, stream>>>(
        x, Wih1, Whh1, bih1, bhh1, Wih2, Whh2, bih2, bhh2, Wout, bout, fut,
        out, h1buf, h2buf, partials, cnt);
}